// GritMessagePassing_42863773614509
// MI455X (gfx1250) — compile-verified
//
#include <hip/hip_runtime.h>
#include <math.h>

// GRIT message passing for MI455X (gfx1250, wave32).
// All GEMMs (K=64, Nout=64) run on V_WMMA_F32_16X16X4_F32 (exact fp32).
//
// Fragment layout assumptions (CDNA5 ISA ch 7.12.2, wave32):
//   A 16x4 f32 : lane l, vgpr v -> A[M = l%16][K = 2*(l/16) + v]
//   B 4x16 f32 : lane l, vgpr v -> B[K = 2*(l/16) + v][N = l%16]   (assumed, symmetric to A)
//   C/D 16x16  : lane l, vgpr v -> D[M = v + 8*(l/16)][N = l%16]

#define NN 50000
#define NE 800000
#define HD 64          // H*D = IN_DIM = 64
#define LP 68          // padded LDS row stride (floats): conflict-free v2f reads

typedef float v2f __attribute__((ext_vector_type(2)));
typedef float v8f __attribute__((ext_vector_type(8)));

__device__ __forceinline__ v8f wmma_f32(v2f a, v2f b, v8f c) {
  return __builtin_amdgcn_wmma_f32_16x16x4_f32(false, a, false, b, (short)0, c, false, false);
}

// Stage a 16x64 fp32 tile (row-major, contiguous) into padded LDS using the
// CDNA5 async-to-LDS engine: 8 x global_load_async_to_lds_b128 per lane,
// tracked on ASYNCcnt (no VGPR round trip). GVS addressing: saddr = tile base
// (wave-uniform -> SGPR pair), vaddr = per-lane byte offset.
__device__ __forceinline__ void stage_tile_async(const float* g, const float* s, int lane) {
  unsigned lbase = (unsigned)(size_t)s;   // low 32 bits of generic ptr = LDS byte offset
#pragma unroll
  for (int j = 0; j < 8; ++j) {
    int idx = lane + 32 * j;              // 16B-granule index, 0..255 (16 rows x 16)
    int row = idx >> 4, c4 = idx & 15;
    unsigned ldsa = lbase + (unsigned)((row * LP + c4 * 4) * 4);
    unsigned voff = (unsigned)(idx * 16);
    asm volatile("global_load_async_to_lds_b128 %0, %1, %2"
                 :: "v"(ldsa), "v"(voff), "s"(g)
                 : "memory");
  }
  asm volatile("s_wait_asynccnt 0" ::: "memory");
}

// One 16x64 @ 64x64^T GEMM tile: preload B fragments (one clause), then 16
// back-to-back WMMAs per 16x16 output tile; store D to O (global).
__device__ __forceinline__ void gemm_store(const v2f afr[16], const float* __restrict__ W,
                                           float* __restrict__ O, int tile, int lane) {
  int n = lane & 15, kh = (lane >> 4) << 1, mbase = tile * 16 + 8 * (lane >> 4);
  for (int nt = 0; nt < 4; ++nt) {
    v2f bfr[16];
#pragma unroll
    for (int kt = 0; kt < 16; ++kt)
      bfr[kt] = *(const v2f*)(W + (nt * 16 + n) * HD + kt * 4 + kh);
    v8f acc = {};
#pragma unroll
    for (int kt = 0; kt < 16; ++kt) acc = wmma_f32(afr[kt], bfr[kt], acc);
    int f = nt * 16 + n;
#pragma unroll
    for (int v = 0; v < 8; ++v) O[(size_t)(mbase + v) * HD + f] = acc[v];
  }
}

// ---------------- Kernel 1: Qh/Kh/Vh = x @ W{Q,K,V}^T ----------------
__global__ __launch_bounds__(128) void proj_nodes(
    const float* __restrict__ x,
    const float* __restrict__ WQ, const float* __restrict__ WK, const float* __restrict__ WV,
    float* __restrict__ Qh, float* __restrict__ Kh, float* __restrict__ Vh) {
  __shared__ __align__(16) float xs[4][16 * LP];
  int wave = threadIdx.x >> 5, lane = threadIdx.x & 31;
  int tile = blockIdx.x * 4 + wave;               // 16-node tile
  bool active = tile < (NN / 16);
  float* s = xs[wave];
  if (active) stage_tile_async(x + (size_t)tile * 16 * HD, s, lane);
  __syncthreads();
  if (!active) return;

  v2f afr[16];
  int m = lane & 15, kh = (lane >> 4) << 1;
#pragma unroll
  for (int kt = 0; kt < 16; ++kt)
    afr[kt] = *(const v2f*)(s + m * LP + kt * 4 + kh);

  gemm_store(afr, WQ, Qh, tile, lane);   // direct kernarg pointers per call:
  gemm_store(afr, WK, Kh, tile, lane);   // address-space inference -> global_*
  gemm_store(afr, WV, Vh, tile, lane);
}

// ---------------- Kernel 2: fused edge pipeline ----------------
// Ew,Eb GEMMs -> conn = sign*sqrt((Qh[dst]+Kh[src])*Ew) + Eb -> Eo GEMM (to d_out)
// -> score = conn . Aw, clamp, exp, atomicAdd(den).
__global__ __launch_bounds__(128) void edge_fused(
    const float* __restrict__ ea, const int* __restrict__ eidx,
    const float* __restrict__ WEw, const float* __restrict__ WEb, const float* __restrict__ bEb,
    const float* __restrict__ WEo, const float* __restrict__ bEo, const float* __restrict__ Aw,
    const float* __restrict__ Qh, const float* __restrict__ Kh,
    float* __restrict__ EoOut, float* __restrict__ sexp, float* __restrict__ den) {
  __shared__ __align__(16) float es_all[4][16 * LP];
  __shared__ __align__(16) float cs_all[4][16 * LP];
  __shared__ int idx_all[4][32];
  int wave = threadIdx.x >> 5, lane = threadIdx.x & 31;
  int tile = blockIdx.x * 4 + wave;               // 16-edge tile, grid is exact
  float* es = es_all[wave];
  float* cs = cs_all[wave];
  int* idx = idx_all[wave];

  // pull the next block's tile toward L2 while we work on this one
  if ((size_t)(tile + 4) < NE / 16) __builtin_prefetch(ea + (size_t)(tile + 4) * 16 * HD, 0, 1);

  stage_tile_async(ea + (size_t)tile * 16 * HD, es, lane);
  if (lane < 16) idx[lane] = eidx[tile * 16 + lane];                 // dst
  else           idx[lane] = eidx[NE + tile * 16 + (lane - 16)];     // src
  __syncthreads();

  int m = lane & 15, kh = (lane >> 4) << 1, n = lane & 15, mhi = 8 * (lane >> 4);
  v2f afr[16];
#pragma unroll
  for (int kt = 0; kt < 16; ++kt)
    afr[kt] = *(const v2f*)(es + m * LP + kt * 4 + kh);

  for (int nt = 0; nt < 4; ++nt) {
    v8f accw = {}, accb = {};
    {
      v2f bfr[16];
#pragma unroll
      for (int kt = 0; kt < 16; ++kt)
        bfr[kt] = *(const v2f*)(WEw + (nt * 16 + n) * HD + kt * 4 + kh);
#pragma unroll
      for (int kt = 0; kt < 16; ++kt) accw = wmma_f32(afr[kt], bfr[kt], accw);
    }
    {
      v2f bfr[16];
#pragma unroll
      for (int kt = 0; kt < 16; ++kt)
        bfr[kt] = *(const v2f*)(WEb + (nt * 16 + n) * HD + kt * 4 + kh);
#pragma unroll
      for (int kt = 0; kt < 16; ++kt) accb = wmma_f32(afr[kt], bfr[kt], accb);
    }
    int f = nt * 16 + n;
    float bEbf = bEb[f];
#pragma unroll
    for (int v = 0; v < 8; ++v) {
      int el = v + mhi;
      float q = Qh[(size_t)idx[el] * HD + f];
      float k = Kh[(size_t)idx[16 + el] * HD + f];
      float c1 = (q + k) * accw[v];
      float c2 = copysignf(sqrtf(fabsf(c1)), c1);   // sign(x)*sqrt(|x|)
      cs[el * LP + f] = c2 + accb[v] + bEbf;
    }
  }
  __syncthreads();

  // conn now in LDS -> re-fragment as A operand for the Eo GEMM
  v2f cfr[16];
#pragma unroll
  for (int kt = 0; kt < 16; ++kt)
    cfr[kt] = *(const v2f*)(cs + m * LP + kt * 4 + kh);

  for (int nt = 0; nt < 4; ++nt) {
    v2f bfr[16];
#pragma unroll
    for (int kt = 0; kt < 16; ++kt)
      bfr[kt] = *(const v2f*)(WEo + (nt * 16 + n) * HD + kt * 4 + kh);
    v8f acc = {};
#pragma unroll
    for (int kt = 0; kt < 16; ++kt) acc = wmma_f32(cfr[kt], bfr[kt], acc);
    int f = nt * 16 + n;
    float bo = bEo[f];
#pragma unroll
    for (int v = 0; v < 8; ++v)
      EoOut[(size_t)(tile * 16 + mhi + v) * HD + f] = acc[v] + bo;
  }

  // scores: 16 edges x 8 heads = 128 pairs -> 4 per lane.
  // clamp(+-5) before softmax => exp(score) is bounded, no max-subtraction needed.
#pragma unroll
  for (int i = 0; i < 4; ++i) {
    int p = lane + 32 * i;
    int el = p >> 3, h = p & 7;
    float sc = 0.f;
#pragma unroll
    for (int d = 0; d < 8; ++d) sc += cs[el * LP + h * 8 + d] * Aw[d * 8 + h];
    sc = fminf(fmaxf(sc, -5.0f), 5.0f);
    float ex = __expf(sc);
    sexp[(size_t)(tile * 16 + el) * 8 + h] = ex;
    atomicAdd(&den[(size_t)idx[el] * 8 + h], ex);
  }
}

// ---------------- Kernel 3: prob = exp/den; scatter V*prob and Eo*prob ----------------
__global__ __launch_bounds__(256) void agg_kernel(
    const int* __restrict__ eidx, const float* __restrict__ Vh,
    const float* __restrict__ Eo, const float* __restrict__ sexp, const float* __restrict__ den,
    float* __restrict__ agg, float* __restrict__ rowAcc) {
  int t = blockIdx.x * 256 + threadIdx.x;
  if (t >= NE * 8) return;
  int e = t >> 3, h = t & 7;
  int de = eidx[e], se = eidx[NE + e];
  float prob = sexp[t] / (den[(size_t)de * 8 + h] + 1e-16f);
#pragma unroll
  for (int d = 0; d < 8; ++d) {
    atomicAdd(&agg[(size_t)de * HD + h * 8 + d],    Vh[(size_t)se * HD + h * 8 + d] * prob);
    atomicAdd(&rowAcc[(size_t)de * HD + h * 8 + d], Eo[(size_t)e * HD + h * 8 + d] * prob);
  }
}

// ---------------- Kernel 4: h_out = agg + rowAcc . BW ----------------
__global__ __launch_bounds__(256) void finalize_kernel(
    const float* __restrict__ agg, const float* __restrict__ rowAcc,
    const float* __restrict__ BW, float* __restrict__ hout) {
  int t = blockIdx.x * 256 + threadIdx.x;
  if (t >= NN * HD) return;
  int n = t >> 6, hc = t & 63, h = hc >> 3, c = hc & 7;
  float s = 0.f;
#pragma unroll
  for (int d = 0; d < 8; ++d)
    s += rowAcc[(size_t)n * HD + h * 8 + d] * BW[(d * 8 + h) * 8 + c];
  hout[t] = agg[t] + s;
}

extern "C" void kernel_launch(void* const* d_in, const int* in_sizes, int n_in,
                              void* d_out, int out_size, void* d_ws, size_t ws_size,
                              hipStream_t stream) {
  const float* x   = (const float*)d_in[0];
  const float* ea  = (const float*)d_in[1];
  const int*   ei  = (const int*)d_in[2];
  const float* WQ  = (const float*)d_in[3];
  const float* WK  = (const float*)d_in[4];
  const float* WV  = (const float*)d_in[5];
  const float* WEw = (const float*)d_in[6];
  const float* WEb = (const float*)d_in[7];
  const float* bEb = (const float*)d_in[8];
  const float* WEo = (const float*)d_in[9];
  const float* bEo = (const float*)d_in[10];
  const float* Aw  = (const float*)d_in[11];
  const float* BW  = (const float*)d_in[12];

  float* out   = (float*)d_out;
  float* hout  = out;                       // [N, 64]
  float* EoOut = out + (size_t)NN * HD;     // [E, 64]

  // Workspace layout (~91 MB of fp32):
  float* ws   = (float*)d_ws;
  float* Qh   = ws;                          // N*64
  float* Kh   = Qh + (size_t)NN * HD;        // N*64
  float* Vh   = Kh + (size_t)NN * HD;        // N*64
  float* sexp = Vh + (size_t)NN * HD;        // E*8
  float* den  = sexp + (size_t)NE * 8;       // N*8   (zeroed)
  float* agg  = den + (size_t)NN * 8;        // N*64  (zeroed)
  float* rAcc = agg + (size_t)NN * HD;       // N*64  (zeroed)

  hipMemsetAsync(den, 0, (size_t)(NN * 8 + NN * HD + NN * HD) * sizeof(float), stream);

  proj_nodes<<<(NN / 16 + 3) / 4, 128, 0, stream>>>(x, WQ, WK, WV, Qh, Kh, Vh);
  edge_fused<<<NE / 16 / 4, 128, 0, stream>>>(ea, ei, WEw, WEb, bEb, WEo, bEo, Aw,
                                              Qh, Kh, EoOut, sexp, den);
  agg_kernel<<<(NE * 8) / 256, 256, 0, stream>>>(ei, Vh, EoOut, sexp, den, agg, rAcc);
  finalize_kernel<<<(NN * HD) / 256, 256, 0, stream>>>(agg, rAcc, BW, hout);
}